// SPGG_85504208929338
// MI455X (gfx1250) — compile-verified
//
#include <hip/hip_runtime.h>
#include <stdint.h>

// SPGG Fermi update, fused single pass.
// L=4096 torus, R=5 (so R/5 == 1), K=0.1 (so 1/K == 10).
//
// profit = 2*ns + stencil4(ns) - 6*state, where ns = 5pt-stencil(state).
// out    = (p <= sigmoid((profit[nbr]-profit)/K)) ? state[nbr] : state,
// with nbr chosen by learning_direction (0=left,1=right,2=up,3=down).
//
// Tile layout (per 256-thread block, 64x64 outputs):
//   s_state : 70 rows x 72 cols, global origin (by*64-3, bx*64-4)
//             -> output (r,c) maps to s_state[r+3][c+4]
//   s_ns    : 68x68 at centers s_state[r+1][c+2]
//   s_profit: 66x66 at centers s_ns[r+1][c+1], state s_state[r+2][c+3]

#define LDIM   4096
#define LMASK  4095
#define TILE   64
#define NTHR   256
#define SH     70   // state rows (halo 3 top/bottom)
#define SW     72   // state cols (halo 4 left/right)
#define SS     72   // LDS stride (floats)
#define NH     68
#define NS     72
#define PH     66
#define PS     72
#define NSEG   18   // 72 cols / 4 = 18 B128 segments per row

// CDNA5 async global->LDS loads (ASYNCcnt-tracked, bypass VGPRs).
__device__ __forceinline__ void async_ld_b32(uint32_t lds_off, uint64_t gaddr) {
    asm volatile("global_load_async_to_lds_b32 %0, %1, off"
                 :: "v"(lds_off), "v"(gaddr) : "memory");
}
__device__ __forceinline__ void async_ld_b128(uint32_t lds_off, uint64_t gaddr) {
    asm volatile("global_load_async_to_lds_b128 %0, %1, off"
                 :: "v"(lds_off), "v"(gaddr) : "memory");
}

__device__ __forceinline__ float fermi_cell(const float* s_state, const float* s_profit,
                                            int r, int c, int d, float p) {
    // direction -> (dr,dc): 0=left(0,-1) 1=right(0,+1) 2=up(-1,0) 3=down(+1,0)
    int dr = (d == 2) ? -1 : ((d == 3) ? 1 : 0);
    int dc = (d == 0) ? -1 : ((d == 1) ? 1 : 0);
    float pc = s_profit[(r + 1) * PS + (c + 1)];
    float pn = s_profit[(r + 1 + dr) * PS + (c + 1 + dc)];
    float sc = s_state[(r + 3) * SS + (c + 4)];
    float sn = s_state[(r + 3 + dr) * SS + (c + 4 + dc)];
    // W = sigmoid((pn - pc)/K) = 1 / (1 + exp((pc - pn)*10))
    float e = __expf((pc - pn) * 10.0f);
    float W = __builtin_amdgcn_rcpf(1.0f + e);
    return (p <= W) ? sn : sc;
}

__global__ __launch_bounds__(NTHR)
void spgg_fermi_kernel(const float* __restrict__ state,
                       const int*   __restrict__ ldir,
                       const float* __restrict__ prob,
                       float*       __restrict__ out)
{
    __shared__ float s_state[SH * SS];
    __shared__ float s_ns[NH * NS];
    __shared__ float s_profit[PH * PS];

    const int tid  = threadIdx.x;
    const int bx   = blockIdx.x;
    const int by   = blockIdx.y;
    const int row0 = by * TILE - 3;      // global row of s_state row 0
    const int cols = bx * TILE - 4;      // global col of s_state col 0

    // ---- Phase A: async-stage state tile into LDS ------------------------
    if (bx >= 1 && bx <= (LDIM / TILE - 2)) {
        // Interior block column: rows are contiguous in memory,
        // 16B-aligned both in global (cols % 4 == 0) and LDS (stride 288B).
        for (int t = tid; t < SH * NSEG; t += NTHR) {
            int r = t / NSEG;
            int s = t - r * NSEG;
            int gr = (row0 + r) & LMASK;             // vertical torus wrap
            uint32_t lo = (uint32_t)(uintptr_t)(&s_state[r * SS + s * 4]);
            uint64_t ga = (uint64_t)(uintptr_t)(state + (((size_t)gr) << 12) + cols + s * 4);
            async_ld_b128(lo, ga);
        }
    } else {
        // Edge block columns (bx==0 or bx==63): B128 would straddle the
        // horizontal wrap, so load scalars with wrapped columns.
        for (int t = tid; t < SH * SW; t += NTHR) {
            int r = t / SW;
            int c = t - r * SW;
            int gr = (row0 + r) & LMASK;
            int gc = (cols + c) & LMASK;
            uint32_t lo = (uint32_t)(uintptr_t)(&s_state[r * SS + c]);
            uint64_t ga = (uint64_t)(uintptr_t)(state + (((size_t)gr) << 12) + gc);
            async_ld_b32(lo, ga);
        }
    }
    asm volatile("s_wait_asynccnt 0x0" ::: "memory");
    __syncthreads();

    // ---- Phase B: ns = 5-point stencil of state, 68x68 -------------------
    for (int t = tid; t < NH * NH; t += NTHR) {
        int r = t / NH;
        int c = t - r * NH;
        const float* p0 = &s_state[(r + 1) * SS + (c + 2)];
        s_ns[r * NS + c] = p0[0] + p0[-SS] + p0[SS] + p0[-1] + p0[1];
    }
    __syncthreads();

    // ---- Phase C: profit = 2*ns + stencil4(ns) - 6*state, 66x66 ----------
    for (int t = tid; t < PH * PH; t += NTHR) {
        int r = t / PH;
        int c = t - r * PH;
        const float* q = &s_ns[(r + 1) * NS + (c + 1)];
        float acc = 2.0f * q[0] + q[-NS] + q[NS] + q[-1] + q[1];
        float st  = s_state[(r + 2) * SS + (c + 3)];
        s_profit[r * PS + c] = acc - 6.0f * st;   // (R/5)==1 folded
    }
    __syncthreads();

    // ---- Phase D: Fermi imitation, 64x64 outputs, 4-wide vmem ------------
    for (int t = tid; t < (TILE * TILE) / 4; t += NTHR) {
        int r = t >> 4;
        int c = (t & 15) << 2;
        size_t g = (((size_t)(by * TILE + r)) << 12) + (size_t)(bx * TILE + c);

        const int4   d4 = *reinterpret_cast<const int4*>(ldir + g);
        const float4 p4 = *reinterpret_cast<const float4*>(prob + g);
        float4 o;
        o.x = fermi_cell(s_state, s_profit, r, c + 0, d4.x, p4.x);
        o.y = fermi_cell(s_state, s_profit, r, c + 1, d4.y, p4.y);
        o.z = fermi_cell(s_state, s_profit, r, c + 2, d4.z, p4.z);
        o.w = fermi_cell(s_state, s_profit, r, c + 3, d4.w, p4.w);
        *reinterpret_cast<float4*>(out + g) = o;
    }
}

extern "C" void kernel_launch(void* const* d_in, const int* in_sizes, int n_in,
                              void* d_out, int out_size, void* d_ws, size_t ws_size,
                              hipStream_t stream) {
    (void)in_sizes; (void)n_in; (void)d_ws; (void)ws_size; (void)out_size;
    const float* state = (const float*)d_in[0];
    const int*   ldir  = (const int*)d_in[1];
    const float* prob  = (const float*)d_in[2];
    float*       out   = (float*)d_out;

    dim3 grid(LDIM / TILE, LDIM / TILE);   // 64 x 64 blocks
    spgg_fermi_kernel<<<grid, NTHR, 0, stream>>>(state, ldir, prob, out);
}